// FilterDetections_31361851195597
// MI455X (gfx1250) — compile-verified
//
#include <hip/hip_runtime.h>
#include <cstdint>

// ---------------- problem constants (match reference) ----------------
namespace {
constexpr int   kB       = 2;
constexpr int   kN       = 20000;
constexpr int   kC       = 20;
constexpr int   kMaxDet  = 300;
constexpr float kScoreThr = 0.05f;
constexpr float kNmsThr   = 0.5f;
constexpr float kNeg      = -1000000000.0f;   // NEG
constexpr float kSent     = -3.4e38f;         // below any real/NEG score

constexpr int kT1      = 1024;                // 32 wave32 waves
constexpr int kItems1  = 20;                  // 1024*20 = 20480 >= 20000
constexpr int kBoxLds  = kN * 16;             // 312.5 KB of the 320 KB WGP LDS
constexpr int kSmem1   = kBoxLds + 256;       // + reduction scratch

constexpr int kR       = kC * kMaxDet;        // 6000 candidates per image
constexpr int kT2      = 1024;
constexpr int kItems2  = 6;                   // 1024*6 >= 6000
}

// Block-wide argmax with lowest-index tie-break (matches jnp.argmax / lax.top_k).
// Requires exactly 32 waves per block. Broadcasts result to all threads.
__device__ __forceinline__ void block_argmax(float &bv, int &bi, float *pV, int *pI) {
  const int tid = threadIdx.x;
#pragma unroll
  for (int off = 16; off > 0; off >>= 1) {
    float ov = __shfl_xor(bv, off, 32);
    int   oi = __shfl_xor(bi, off, 32);
    if (ov > bv || (ov == bv && oi < bi)) { bv = ov; bi = oi; }
  }
  const int wave = tid >> 5;
  const int lane = tid & 31;
  if (lane == 0) { pV[wave] = bv; pI[wave] = bi; }
  __syncthreads();
  if (wave == 0) {
    float v = pV[lane];
    int   i = pI[lane];
#pragma unroll
    for (int off = 16; off > 0; off >>= 1) {
      float ov = __shfl_xor(v, off, 32);
      int   oi = __shfl_xor(i, off, 32);
      if (ov > v || (ov == v && oi < i)) { v = ov; i = oi; }
    }
    if (lane == 0) { pV[0] = v; pI[0] = i; }
  }
  __syncthreads();
  bv = pV[0];
  bi = pI[0];
  __syncthreads();   // protect scratch before next round's writes
}

// ---------------- Phase 1: per-(image,class) greedy NMS ----------------
__global__ __launch_bounds__(kT1) void nms_class_kernel(
    const float* __restrict__ boxesG, const float* __restrict__ clsG,
    float* __restrict__ scoreWs, int* __restrict__ idxWs, int* __restrict__ validWs) {
  extern __shared__ __align__(16) char smem[];
  float4* bl4 = (float4*)smem;                       // boxes tile in LDS
  float*  pV  = (float*)(smem + kBoxLds);
  int*    pI  = (int*)(smem + kBoxLds + 128);

  const int tid = threadIdx.x;
  const int b   = blockIdx.x / kC;
  const int c   = blockIdx.x % kC;

  // --- CDNA5 async copy: stage this image's 312.5 KB box tensor global -> LDS ---
  const unsigned ldsBase = (unsigned)__builtin_amdgcn_groupstaticsize(); // dynamic-LDS start
  const char* gbox = (const char*)(boxesG + (size_t)b * kN * 4);
#pragma unroll
  for (int k = 0; k < kItems1; ++k) {
    int idx = tid + (k << 10);
    if (idx < kN) {
      unsigned dst = ldsBase + (unsigned)idx * 16u;
      uint64_t src = (uint64_t)gbox + (uint64_t)idx * 16u;
      asm volatile("global_load_async_to_lds_b128 %0, %1, off"
                   :: "v"(dst), "v"(src) : "memory");
    }
  }

  // --- thresholded scores into registers (overlaps the async staging) ---
  float sc[kItems1];
#pragma unroll
  for (int k = 0; k < kItems1; ++k) {
    int idx = tid + (k << 10);
    float s = kNeg;
    if (idx < kN) {
      float v = clsG[((size_t)b * kN + idx) * kC + c];
      s = (v > kScoreThr) ? v : kNeg;
    }
    sc[k] = s;
  }

  asm volatile("s_wait_asynccnt 0" ::: "memory");
  __syncthreads();

  const int base = blockIdx.x * kMaxDet;
  int step = 0;
  for (; step < kMaxDet; ++step) {
    // local argmax over registers; ascending-index scan keeps lowest index on ties
    float bv = kSent; int bi = 0x7fffffff;
#pragma unroll
    for (int k = 0; k < kItems1; ++k) {
      int idx = tid + (k << 10);
      if (sc[k] > bv) { bv = sc[k]; bi = idx; }
    }
    block_argmax(bv, bi, pV, pI);

    // once max <= NEG/2, every later step is invalid (scores only decrease)
    if (bv <= kNeg * 0.5f) break;

    if (tid == 0) { scoreWs[base + step] = bv; idxWs[base + step] = bi; validWs[base + step] = 1; }

    // broadcast selected box from LDS; suppress overlapping candidates
    float4 bb = bl4[bi];
    float  ai = (bb.z - bb.x) * (bb.w - bb.y);
#pragma unroll
    for (int k = 0; k < kItems1; ++k) {
      int idx = tid + (k << 10);
      if (idx < kN) {
        float4 oj = bl4[idx];
        float aj  = (oj.z - oj.x) * (oj.w - oj.y);
        float ix1 = fmaxf(bb.x, oj.x);
        float iy1 = fmaxf(bb.y, oj.y);
        float ix2 = fminf(bb.z, oj.z);
        float iy2 = fminf(bb.w, oj.w);
        float inter = fmaxf(ix2 - ix1, 0.0f) * fmaxf(iy2 - iy1, 0.0f);
        float iou   = inter / (ai + aj - inter + 1e-8f);
        if (iou > kNmsThr || idx == bi) sc[k] = kNeg;
      }
    }
  }
  // fill remaining (invalid) slots exactly as the reference's masked outputs need
  for (int t = step + tid; t < kMaxDet; t += kT1) {
    scoreWs[base + t] = kNeg; idxWs[base + t] = 0; validWs[base + t] = 0;
  }
}

// ---------------- Phase 2: per-image top-k over 6000 candidates ----------------
__global__ __launch_bounds__(kT2) void topk_kernel(
    const float* __restrict__ boxesG,
    const float* __restrict__ scoreWs, const int* __restrict__ idxWs,
    const int* __restrict__ validWs,
    float* __restrict__ outBoxes, float* __restrict__ outScores,
    int* __restrict__ outLabels) {
  __shared__ float pV[32];
  __shared__ int   pI[32];
  const int tid = threadIdx.x;
  const int b   = blockIdx.x;

  float v[kItems2];
#pragma unroll
  for (int k = 0; k < kItems2; ++k) {
    int ci = tid + (k << 10);
    v[k] = (ci < kR) ? scoreWs[b * kR + ci] : kSent;
  }

  for (int t = 0; t < kMaxDet; ++t) {
    float bv = kSent; int bi = 0x7fffffff;
#pragma unroll
    for (int k = 0; k < kItems2; ++k) {
      int ci = tid + (k << 10);
      if (v[k] > bv) { bv = v[k]; bi = ci; }
    }
    block_argmax(bv, bi, pV, pI);

    if (tid == 0) {
      int gi     = b * kR + bi;
      int srcIdx = idxWs[gi];
      int val    = validWs[gi];
      int label  = bi / kMaxDet;                    // candidate -> class
      outScores[b * kMaxDet + t] = val ? bv : -1.0f;
      outLabels[b * kMaxDet + t] = val ? label : -1;
      float4 bb = ((const float4*)boxesG)[(size_t)b * kN + srcIdx];
      float4 ob;
      ob.x = val ? bb.x : -1.0f; ob.y = val ? bb.y : -1.0f;
      ob.z = val ? bb.z : -1.0f; ob.w = val ? bb.w : -1.0f;
      ((float4*)outBoxes)[b * kMaxDet + t] = ob;
    }
    // remove the winner (stable top_k: equal values -> lower index first)
#pragma unroll
    for (int k = 0; k < kItems2; ++k) {
      int ci = tid + (k << 10);
      if (ci == bi) v[k] = kSent;
    }
  }
}

// ---------------- host launcher ----------------
extern "C" void kernel_launch(void* const* d_in, const int* in_sizes, int n_in,
                              void* d_out, int out_size, void* d_ws, size_t ws_size,
                              hipStream_t stream) {
  (void)in_sizes; (void)n_in; (void)out_size; (void)ws_size;
  const float* boxes = (const float*)d_in[0];   // [B, N, 4] f32
  const float* cls   = (const float*)d_in[1];   // [B, N, C] f32

  // workspace: 12000 entries each of score/idx/valid (144 KB total)
  float* scoreWs = (float*)d_ws;
  int*   idxWs   = (int*)((char*)d_ws + (size_t)kB * kR * sizeof(float));
  int*   validWs = (int*)((char*)d_ws + (size_t)2 * kB * kR * sizeof(float));

  // d_out: boxes [B,300,4] | scores [B,300] | labels(int32) [B,300]
  float* outBoxes  = (float*)d_out;
  float* outScores = outBoxes + kB * kMaxDet * 4;
  int*   outLabels = (int*)d_out + kB * kMaxDet * 4 + kB * kMaxDet;

  // allow 312.5 KB + scratch of dynamic LDS (WGP has 320 KB)
  (void)hipFuncSetAttribute(reinterpret_cast<const void*>(nms_class_kernel),
                            hipFuncAttributeMaxDynamicSharedMemorySize, kSmem1);

  nms_class_kernel<<<dim3(kB * kC), dim3(kT1), kSmem1, stream>>>(
      boxes, cls, scoreWs, idxWs, validWs);
  topk_kernel<<<dim3(kB), dim3(kT2), 0, stream>>>(
      boxes, scoreWs, idxWs, validWs, outBoxes, outScores, outLabels);
}